// BlockSONAR_Model_24189255811079
// MI455X (gfx1250) — compile-verified
//
#include <hip/hip_runtime.h>
#include <hip/hip_bf16.h>

// ---------------- problem constants (match reference) ----------------
#define NN   50000
#define EE   400000
#define INF_ 128
#define HH   128
#define OUTD 64
#define NB   3
#define NITER 2
#define EPS_ 0.01f

typedef __attribute__((ext_vector_type(2))) float v2f;
typedef __attribute__((ext_vector_type(8))) float v8f;

// =====================================================================
// GEMM: Y[nrows, MT*16] = act( X[nrows,128] @ W[MT*16, 128]^T + bias )
// fp32 WMMA 16x16x4 (keeps reference fp32 precision; problem is
// memory-bound so the fp32 matrix rate is more than sufficient).
//
// Workgroup = 256 threads = 8 waves; wave -> one 16-row x MT*16-col tile.
// W is staged in LDS once per workgroup (row stride padded to 132 floats
// so the 16-row B-fragment reads spread across all 64 LDS banks).
// K-loop fully unrolled (K=128 compile-time) with A + all B fragments
// loaded before the 8 WMMAs of each step, so loads pipeline instead of
// a full s_wait_loadcnt 0 before every WMMA.
// =====================================================================
template <int MT, int ACT>
__global__ __launch_bounds__(256) void k_gemm_wmma(
    const float* __restrict__ X, int ldx,
    const float* __restrict__ W, int ldw,
    const float* __restrict__ bias,
    float* __restrict__ Y, int ldy,
    int nrows)
{
    constexpr int KK   = 128;
    constexpr int LDSW = KK + 4;                 // padded row stride (floats)
    __shared__ float w_lds[MT * 16 * LDSW];

    // ---- cooperative W stage: MT*16 rows x 128 floats, coalesced float4
    {
        constexpr int row_f4 = KK / 4;           // 32 float4 per row
        constexpr int tot_f4 = MT * 16 * row_f4;
        for (int i = threadIdx.x; i < tot_f4; i += 256) {
            const int row = i / row_f4;
            const int c4  = i - row * row_f4;
            const float4 wv = ((const float4*)(W + (size_t)row * ldw))[c4];
            float* dp = &w_lds[row * LDSW + c4 * 4];
            dp[0] = wv.x; dp[1] = wv.y; dp[2] = wv.z; dp[3] = wv.w;
        }
    }
    __syncthreads();

    const int lane = threadIdx.x & 31;
    const int wave = threadIdx.x >> 5;
    const int row0 = (blockIdx.x * 8 + wave) * 16;
    if (row0 >= nrows) return;

    const int half = lane >> 4;                  // 0: lanes 0-15, 1: 16-31
    const int lrow = lane & 15;

    v8f acc[MT];
#pragma unroll
    for (int t = 0; t < MT; ++t) {
        const float bv = bias ? bias[t * 16 + lrow] : 0.0f;
#pragma unroll
        for (int j = 0; j < 8; ++j) acc[t][j] = bv;
    }

    const float* xr = X + (size_t)(row0 + lrow) * ldx + 2 * half;
    const float* wl = &w_lds[lrow * LDSW + 2 * half];

#pragma unroll
    for (int k = 0; k < KK; k += 4) {
        v2f a;
        a.x = xr[k];
        a.y = xr[k + 1];
        v2f b[MT];
#pragma unroll
        for (int t = 0; t < MT; ++t) {
            const float* wp = wl + t * (16 * LDSW) + k;
            b[t].x = wp[0];
            b[t].y = wp[1];
        }
#pragma unroll
        for (int t = 0; t < MT; ++t) {
            acc[t] = __builtin_amdgcn_wmma_f32_16x16x4_f32(
                false, a, false, b[t], (short)0, acc[t], false, false);
        }
    }

#pragma unroll
    for (int t = 0; t < MT; ++t) {
#pragma unroll
        for (int rr = 0; rr < 8; ++rr) {
            float v = acc[t][rr];
            if (ACT == 1) v = fmaxf(v, 0.0f);
            if (ACT == 2) v = tanhf(v);
            const int row = row0 + rr + 8 * half;
            Y[(size_t)row * ldy + t * 16 + lrow] = v;
        }
    }
}

// =====================================================================
// Edge resistance: r[e] = |relu(P[src]+Q[dst]+b1) . w2 + b2|
// and deg[src] += r[e].   One wave per edge, float4 per lane.
// =====================================================================
__global__ __launch_bounds__(256) void k_edge_resist(
    const float* __restrict__ P, const float* __restrict__ Q,
    const float* __restrict__ b1, const float* __restrict__ w2,
    const float* __restrict__ b2p,
    const int* __restrict__ src, const int* __restrict__ dst,
    float* __restrict__ r_out, float* __restrict__ deg, int E)
{
    const int e    = (int)((blockIdx.x * blockDim.x + threadIdx.x) >> 5);
    const int lane = threadIdx.x & 31;
    if (e >= E) return;
    const int s = src[e];
    const int d = dst[e];

    const float4 p  = ((const float4*)(P + (size_t)s * HH))[lane];
    const float4 q  = ((const float4*)(Q + (size_t)d * HH))[lane];
    const float4 bb = ((const float4*)b1)[lane];
    const float4 w  = ((const float4*)w2)[lane];

    const float h0 = fmaxf(p.x + q.x + bb.x, 0.0f);
    const float h1 = fmaxf(p.y + q.y + bb.y, 0.0f);
    const float h2 = fmaxf(p.z + q.z + bb.z, 0.0f);
    const float h3 = fmaxf(p.w + q.w + bb.w, 0.0f);
    float part = h0 * w.x + h1 * w.y + h2 * w.z + h3 * w.w;

#pragma unroll
    for (int off = 16; off > 0; off >>= 1)
        part += __shfl_xor(part, off, 32);

    if (lane == 0) {
        const float rv = fabsf(part + b2p[0]);
        r_out[e] = rv;
        __hip_atomic_fetch_add(&deg[s], rv, __ATOMIC_RELAXED,
                               __HIP_MEMORY_SCOPE_AGENT);
    }
}

// =====================================================================
// conv scatter: agg[dst] += r[e] * f[src]   (one wave per edge)
// =====================================================================
__global__ __launch_bounds__(256) void k_conv_scatter(
    const float* __restrict__ f, const float* __restrict__ r,
    const int* __restrict__ src, const int* __restrict__ dst,
    float* __restrict__ agg, int E)
{
    const int e    = (int)((blockIdx.x * blockDim.x + threadIdx.x) >> 5);
    const int lane = threadIdx.x & 31;
    if (e >= E) return;
    const int s = src[e];
    const int d = dst[e];
    const float rv = r[e];

    const float4 fv = ((const float4*)(f + (size_t)s * HH))[lane];
    float* ap = agg + (size_t)d * HH + lane * 4;
    __hip_atomic_fetch_add(ap + 0, rv * fv.x, __ATOMIC_RELAXED, __HIP_MEMORY_SCOPE_AGENT);
    __hip_atomic_fetch_add(ap + 1, rv * fv.y, __ATOMIC_RELAXED, __HIP_MEMORY_SCOPE_AGENT);
    __hip_atomic_fetch_add(ap + 2, rv * fv.z, __ATOMIC_RELAXED, __HIP_MEMORY_SCOPE_AGENT);
    __hip_atomic_fetch_add(ap + 3, rv * fv.w, __ATOMIC_RELAXED, __HIP_MEMORY_SCOPE_AGENT);
}

// =====================================================================
// leapfrog node update:
//   conv = deg[n]*f - agg ; v -= EPS*(conv + diss*v) ; x += EPS*v
// =====================================================================
__global__ __launch_bounds__(256) void k_node_update(
    float* __restrict__ x, float* __restrict__ v,
    const float* __restrict__ f, const float* __restrict__ agg,
    const float* __restrict__ diss, const float* __restrict__ deg, int NHtot)
{
    const int i = blockIdx.x * blockDim.x + threadIdx.x;
    if (i >= NHtot) return;
    const int node = i >> 7;  // H = 128
    const float conv = deg[node] * f[i] - agg[i];
    float vv = v[i];
    vv = vv - EPS_ * (conv + diss[i] * vv);
    v[i] = vv;
    x[i] = x[i] + EPS_ * vv;
}

// =====================================================================
// host launch
// =====================================================================
extern "C" void kernel_launch(void* const* d_in, const int* in_sizes, int n_in,
                              void* d_out, int out_size, void* d_ws, size_t ws_size,
                              hipStream_t stream)
{
    const float* x_in   = (const float*)d_in[0];
    const int*   ei     = (const int*)  d_in[1];
    const float* emb_w  = (const float*)d_in[2];
    const float* emb_b  = (const float*)d_in[3];
    const float* lap_w  = (const float*)d_in[4];
    const float* er_w1  = (const float*)d_in[5];
    const float* er_b1  = (const float*)d_in[6];
    const float* er_w2  = (const float*)d_in[7];
    const float* er_b2  = (const float*)d_in[8];
    const float* diss_w = (const float*)d_in[9];
    const float* diss_b = (const float*)d_in[10];
    const float* mlp_w1 = (const float*)d_in[11];
    const float* mlp_b1 = (const float*)d_in[12];
    const float* mlp_w2 = (const float*)d_in[13];
    const float* mlp_b2 = (const float*)d_in[14];
    const float* ro_w   = (const float*)d_in[15];
    const float* ro_b   = (const float*)d_in[16];
    float* out = (float*)d_out;

    const int* src = ei;
    const int* dst = ei + EE;

    const size_t NH = (size_t)NN * HH;
    float* ws  = (float*)d_ws;
    float* xw  = ws;            // current node features
    float* vw  = ws + 1 * NH;   // velocity
    float* bA  = ws + 2 * NH;   // P  / f
    float* bB  = ws + 3 * NH;   // Q  / agg
    float* bC  = ws + 4 * NH;   // diss / mlp hidden
    float* rw  = ws + 5 * NH;   // edge resistance [E]
    float* degw = rw + EE;      // degree [N]

    const int tiles       = (NN + 15) / 16;
    const int gemm_blocks = (tiles + 7) / 8;
    const int edge_blocks = (EE + 7) / 8;    // 8 waves (edges) per 256-thr block
    const int upd_blocks  = (int)((NH + 255) / 256);

    // embedding: x = x_in @ emb_w^T + emb_b
    k_gemm_wmma<8, 0><<<gemm_blocks, 256, 0, stream>>>(
        x_in, INF_, emb_w, INF_, emb_b, xw, HH, NN);

    for (int b = 0; b < NB; ++b) {
        const float* w1 = er_w1 + (size_t)b * HH * (2 * HH);
        // P = x @ W1a^T ; Q = x @ W1b^T   (er_w1 split: cols [0,H) / [H,2H))
        k_gemm_wmma<8, 0><<<gemm_blocks, 256, 0, stream>>>(
            xw, HH, w1, 2 * HH, nullptr, bA, HH, NN);
        k_gemm_wmma<8, 0><<<gemm_blocks, 256, 0, stream>>>(
            xw, HH, w1 + HH, 2 * HH, nullptr, bB, HH, NN);

        hipMemsetAsync(degw, 0, NN * sizeof(float), stream);
        hipMemsetAsync(vw, 0, NH * sizeof(float), stream);

        k_edge_resist<<<edge_blocks, 256, 0, stream>>>(
            bA, bB, er_b1 + (size_t)b * HH, er_w2 + (size_t)b * HH,
            er_b2 + b, src, dst, rw, degw, EE);

        for (int it = 0; it < NITER; ++it) {
            // f = x @ lap_w^T ; diss = relu(x @ diss_w^T + diss_b)
            k_gemm_wmma<8, 0><<<gemm_blocks, 256, 0, stream>>>(
                xw, HH, lap_w + (size_t)b * HH * HH, HH, nullptr, bA, HH, NN);
            k_gemm_wmma<8, 1><<<gemm_blocks, 256, 0, stream>>>(
                xw, HH, diss_w + (size_t)b * HH * HH, HH,
                diss_b + (size_t)b * HH, bC, HH, NN);

            hipMemsetAsync(bB, 0, NH * sizeof(float), stream);
            k_conv_scatter<<<edge_blocks, 256, 0, stream>>>(
                bA, rw, src, dst, bB, EE);
            k_node_update<<<upd_blocks, 256, 0, stream>>>(
                xw, vw, bA, bB, bC, degw, (int)NH);
        }

        // x = tanh(x @ mlp_w1^T + b1) @ mlp_w2^T + b2
        k_gemm_wmma<8, 2><<<gemm_blocks, 256, 0, stream>>>(
            xw, HH, mlp_w1 + (size_t)b * HH * HH, HH,
            mlp_b1 + (size_t)b * HH, bC, HH, NN);
        k_gemm_wmma<8, 0><<<gemm_blocks, 256, 0, stream>>>(
            bC, HH, mlp_w2 + (size_t)b * HH * HH, HH,
            mlp_b2 + (size_t)b * HH, xw, HH, NN);
    }

    // readout: out = x @ ro_w^T + ro_b   (OUT = 64 -> MT = 4)
    k_gemm_wmma<4, 0><<<gemm_blocks, 256, 0, stream>>>(
        xw, HH, ro_w, HH, ro_b, out, OUTD, NN);
}